// DGHANlayer_28183575396969
// MI455X (gfx1250) — compile-verified
//
#include <hip/hip_runtime.h>
#include <math.h>

typedef __attribute__((ext_vector_type(2))) float v2f;
typedef __attribute__((ext_vector_type(8))) float v8f;

#define NEG_SLOPE 0.2f

// Exact float atomic-max via integer ordering trick (valid incl. -inf init).
__device__ __forceinline__ void atomicMaxFloat(float* addr, float val) {
  if (val >= 0.0f)
    atomicMax((int*)addr, __float_as_int(val));
  else
    atomicMin((unsigned int*)addr, __float_as_uint(val));
}

// ---------------- init: zero accumulators / denom, emax = -inf ----------------
__global__ void init_kernel(float* __restrict__ acc_pc, float* __restrict__ acc_mc,
                            float* __restrict__ emax, float* __restrict__ denom,
                            int accN, int edgN) {
  int i = blockIdx.x * blockDim.x + threadIdx.x;
  if (i < accN) { acc_pc[i] = 0.0f; acc_mc[i] = 0.0f; }
  if (i < edgN) { emax[i] = -__builtin_inff(); denom[i] = 0.0f; }
}

// ---------------- GEMM: h = x @ W (both convs), fp32 WMMA 16x16x4 ----------------
// Block = 256 threads = 8 waves; block handles a 16-row stripe of x (staged in LDS),
// wave w owns output columns [16w, 16w+16). A fragment shared across pc & mc.
__global__ __launch_bounds__(256)
void gemm_kernel(const float* __restrict__ x, const float* __restrict__ Wpc,
                 const float* __restrict__ Wmc, float* __restrict__ hpc,
                 float* __restrict__ hmc) {
  __shared__ float xs[16 * 128];
  const int rowbase = blockIdx.x * 16;
  for (int i = threadIdx.x; i < 16 * 128; i += 256)
    xs[i] = x[(size_t)rowbase * 128 + i];
  __syncthreads();

  const int lane   = threadIdx.x & 31;
  const int wave   = threadIdx.x >> 5;
  const int colbase = wave * 16;
  const int mn    = lane & 15;          // A row / B col
  const int khalf = (lane >> 4) << 1;   // lanes 16-31 carry K+2/K+3

  v8f cpc = {};
  v8f cmc = {};
  for (int k0 = 0; k0 < 128; k0 += 4) {
    v2f a, bpc, bmc;
    const float* ap = &xs[mn * 128 + k0 + khalf];
    a.x = ap[0];
    a.y = ap[1];
    const int wi = (k0 + khalf) * 128 + colbase + mn;
    bpc.x = Wpc[wi]; bpc.y = Wpc[wi + 128];
    bmc.x = Wmc[wi]; bmc.y = Wmc[wi + 128];
    cpc = __builtin_amdgcn_wmma_f32_16x16x4_f32(false, a, false, bpc, (short)0, cpc, false, false);
    cmc = __builtin_amdgcn_wmma_f32_16x16x4_f32(false, a, false, bmc, (short)0, cmc, false, false);
  }
  // C/D layout: VGPR r -> M = r + 8*(lane>=16), N = lane&15
  const int crow = rowbase + ((lane >> 4) << 3);
  const int ccol = colbase + mn;
#pragma unroll
  for (int r = 0; r < 8; ++r) {
    hpc[(size_t)(crow + r) * 128 + ccol] = cpc[r];
    hmc[(size_t)(crow + r) * 128 + ccol] = cmc[r];
  }
}

// ---------------- per-node attention scores: s = <h[n,head,:], a[head,:]> ----------------
// warp per node; lanes 0-15 -> head 0 channels, lanes 16-31 -> head 1.
__global__ __launch_bounds__(256)
void score_kernel(const float* __restrict__ h, const float* __restrict__ a_src,
                  const float* __restrict__ a_dst, float* __restrict__ s_src,
                  float* __restrict__ s_dst, int n) {
  int warp = (blockIdx.x * blockDim.x + threadIdx.x) >> 5;
  int lane = threadIdx.x & 31;
  if (warp >= n) return;
  const float* hn = h + (size_t)warp * 128;
  float ss = 0.0f, sd = 0.0f;
#pragma unroll
  for (int j = 0; j < 4; ++j) {
    int c = lane * 4 + j;            // lanes 0-15 -> c in [0,64), lanes 16-31 -> [64,128)
    float hv = hn[c];
    ss += hv * a_src[c];
    sd += hv * a_dst[c];
  }
#pragma unroll
  for (int m = 1; m < 16; m <<= 1) {  // reduce within each 16-lane half
    ss += __shfl_xor(ss, m, 32);
    sd += __shfl_xor(sd, m, 32);
  }
  if ((lane & 15) == 0) {
    int head = lane >> 4;
    s_src[warp * 2 + head] = ss;
    s_dst[warp * 2 + head] = sd;
  }
}

// ---------------- edge pass 1: e = leaky_relu(s_src[src]+s_dst[dst]); segment max ----------------
__global__ __launch_bounds__(256)
void edge_max_kernel(const int* __restrict__ ei, int E, int n,
                     const float* __restrict__ s_src, const float* __restrict__ s_dst,
                     float* __restrict__ e_store, float* __restrict__ emax) {
  int e = blockIdx.x * blockDim.x + threadIdx.x;
  if (e >= E + n) return;                       // trailing n = self-loops
  int src = (e < E) ? ei[e]     : (e - E);
  int dst = (e < E) ? ei[E + e] : (e - E);
#pragma unroll
  for (int h = 0; h < 2; ++h) {
    float v = s_src[src * 2 + h] + s_dst[dst * 2 + h];
    v = (v > 0.0f) ? v : NEG_SLOPE * v;         // leaky_relu
    e_store[(size_t)e * 2 + h] = v;
    atomicMaxFloat(&emax[dst * 2 + h], v);
  }
}

// ---------------- edge pass 2: e_exp = exp(e - max); segment sum ----------------
__global__ __launch_bounds__(256)
void edge_exp_kernel(const int* __restrict__ ei, int E, int n,
                     float* __restrict__ e_store, const float* __restrict__ emax,
                     float* __restrict__ denom) {
  int e = blockIdx.x * blockDim.x + threadIdx.x;
  if (e >= E + n) return;
  int dst = (e < E) ? ei[E + e] : (e - E);
#pragma unroll
  for (int h = 0; h < 2; ++h) {
    float ex = expf(e_store[(size_t)e * 2 + h] - emax[dst * 2 + h]);
    e_store[(size_t)e * 2 + h] = ex;
    atomicAdd(&denom[dst * 2 + h], ex);
  }
}

// ---------------- edge pass 3: acc[dst] += alpha * h[src] (warp per edge) ----------------
__global__ __launch_bounds__(256)
void edge_acc_kernel(const int* __restrict__ ei, int E, int n,
                     const float* __restrict__ e_store, const float* __restrict__ denom,
                     const float* __restrict__ h, float* __restrict__ acc) {
  int warp = (blockIdx.x * blockDim.x + threadIdx.x) >> 5;
  int lane = threadIdx.x & 31;
  if (warp >= E + n) return;
  int src = (warp < E) ? ei[warp]     : (warp - E);
  int dst = (warp < E) ? ei[E + warp] : (warp - E);
  float a0 = e_store[(size_t)warp * 2 + 0] / denom[dst * 2 + 0];
  float a1 = e_store[(size_t)warp * 2 + 1] / denom[dst * 2 + 1];
  const float* hs = h   + (size_t)src * 128;
  float*       ad = acc + (size_t)dst * 128;
#pragma unroll
  for (int j = 0; j < 4; ++j) {
    int c = lane + 32 * j;                      // c<64 -> head 0, else head 1
    float alpha = (c < 64) ? a0 : a1;
    atomicAdd(&ad[c], alpha * hs[c]);
  }
}

// ---------------- finalize: 0.5*(elu(acc_pc+b_pc) + elu(acc_mc+b_mc)) ----------------
__global__ __launch_bounds__(256)
void finalize_kernel(const float* __restrict__ acc_pc, const float* __restrict__ acc_mc,
                     const float* __restrict__ b_pc, const float* __restrict__ b_mc,
                     float* __restrict__ out, int total) {
  int i = blockIdx.x * blockDim.x + threadIdx.x;
  if (i >= total) return;
  int c = i & 127;
  float vp = acc_pc[i] + b_pc[c];
  vp = (vp > 0.0f) ? vp : (expf(vp) - 1.0f);
  float vm = acc_mc[i] + b_mc[c];
  vm = (vm > 0.0f) ? vm : (expf(vm) - 1.0f);
  out[i] = 0.5f * (vp + vm);
}

extern "C" void kernel_launch(void* const* d_in, const int* in_sizes, int n_in,
                              void* d_out, int out_size, void* d_ws, size_t ws_size,
                              hipStream_t stream) {
  const float* x      = (const float*)d_in[0];
  const float* W_pc   = (const float*)d_in[1];
  const float* asr_pc = (const float*)d_in[2];
  const float* ads_pc = (const float*)d_in[3];
  const float* b_pc   = (const float*)d_in[4];
  const float* W_mc   = (const float*)d_in[5];
  const float* asr_mc = (const float*)d_in[6];
  const float* ads_mc = (const float*)d_in[7];
  const float* b_mc   = (const float*)d_in[8];
  const int*   ei_pc  = (const int*)d_in[9];
  const int*   ei_mc  = (const int*)d_in[10];
  float* out = (float*)d_out;

  const int NN  = in_sizes[0] / 128;   // 50000 nodes
  const int Epc = in_sizes[9] / 2;     // 600000
  const int Emc = in_sizes[10] / 2;

  // workspace carve-up (floats)
  float* ws = (float*)d_ws;
  size_t off = 0;
  float* h_pc    = ws + off; off += (size_t)NN * 128;
  float* h_mc    = ws + off; off += (size_t)NN * 128;
  float* acc_pc  = ws + off; off += (size_t)NN * 128;
  float* acc_mc  = ws + off; off += (size_t)NN * 128;
  float* ssrc_pc = ws + off; off += (size_t)NN * 2;
  float* sdst_pc = ws + off; off += (size_t)NN * 2;
  float* ssrc_mc = ws + off; off += (size_t)NN * 2;
  float* sdst_mc = ws + off; off += (size_t)NN * 2;
  float* emax    = ws + off; off += (size_t)NN * 4;   // [pc heads | mc heads]
  float* denom   = ws + off; off += (size_t)NN * 4;
  float* e_pc    = ws + off; off += (size_t)(Epc + NN) * 2;
  float* e_mc    = ws + off; off += (size_t)(Emc + NN) * 2;
  float* emax_pc = emax;            float* emax_mc = emax + (size_t)NN * 2;
  float* den_pc  = denom;           float* den_mc  = denom + (size_t)NN * 2;

  const int accN = NN * 128;
  init_kernel<<<(accN + 255) / 256, 256, 0, stream>>>(acc_pc, acc_mc, emax, denom,
                                                      accN, NN * 4);

  gemm_kernel<<<(NN + 15) / 16, 256, 0, stream>>>(x, W_pc, W_mc, h_pc, h_mc);

  score_kernel<<<(NN + 7) / 8, 256, 0, stream>>>(h_pc, asr_pc, ads_pc, ssrc_pc, sdst_pc, NN);
  score_kernel<<<(NN + 7) / 8, 256, 0, stream>>>(h_mc, asr_mc, ads_mc, ssrc_mc, sdst_mc, NN);

  const int tpc = Epc + NN, tmc = Emc + NN;
  edge_max_kernel<<<(tpc + 255) / 256, 256, 0, stream>>>(ei_pc, Epc, NN, ssrc_pc, sdst_pc, e_pc, emax_pc);
  edge_max_kernel<<<(tmc + 255) / 256, 256, 0, stream>>>(ei_mc, Emc, NN, ssrc_mc, sdst_mc, e_mc, emax_mc);

  edge_exp_kernel<<<(tpc + 255) / 256, 256, 0, stream>>>(ei_pc, Epc, NN, e_pc, emax_pc, den_pc);
  edge_exp_kernel<<<(tmc + 255) / 256, 256, 0, stream>>>(ei_mc, Emc, NN, e_mc, emax_mc, den_mc);

  edge_acc_kernel<<<(tpc + 7) / 8, 256, 0, stream>>>(ei_pc, Epc, NN, e_pc, den_pc, h_pc, acc_pc);
  edge_acc_kernel<<<(tmc + 7) / 8, 256, 0, stream>>>(ei_mc, Emc, NN, e_mc, den_mc, h_mc, acc_mc);

  finalize_kernel<<<(accN + 255) / 256, 256, 0, stream>>>(acc_pc, acc_mc, b_pc, b_mc, out, accN);
}